// TemporalModule_8229157339186
// MI455X (gfx1250) — compile-verified
//
#include <hip/hip_runtime.h>
#include <math.h>

typedef float v2f __attribute__((ext_vector_type(2)));
typedef float v8f __attribute__((ext_vector_type(8)));

#define AS_LD 68     // 64 data + 4 pad floats per A row (bank-conflict-free frag reads)
#define BS_GRP 160   // per k-pair group: 128 data + 32 pad (shifts upper-half lanes by 32 banks)

// D[M x 512] = A'[M x 512] @ op(W)[512 x 512] (+bias), A' rows optionally remapped by rowmap.
// transB=1: D = A @ W^T  (B[k][n] = W[n][k]);  transB=0: D = A @ W.
// Register-double-buffered: next K-chunk is staged into VGPRs while WMMAs run from LDS.
__global__ __launch_bounds__(256) void gemm512_wmma(
    const float* __restrict__ A, const float* __restrict__ W, float* __restrict__ D,
    const int* __restrict__ rowmap, const float* __restrict__ bias, int transB) {
  __shared__ float As[32 * AS_LD];   // 32 rows x 64 k (padded)
  __shared__ float Bs[32 * BS_GRP];  // 32 k-pair groups x 64 n x 2 (padded)
  const int tid  = threadIdx.x;
  const int lane = tid & 31;
  const int wave = tid >> 5;
  const int wm = wave & 1;   // 2 wave-rows of 16
  const int wn = wave >> 1;  // 4 wave-cols of 16
  const int rowBase = blockIdx.x * 32;
  const int colBase = blockIdx.y * 64;

  // ---- hoisted staging addresses (rowmap resolved ONCE, outside the K loop) ----
  const int ra0 = tid >> 4;            // A rows handled by this thread: ra0, ra0+16
  const int ra1 = ra0 + 16;
  const int ac  = (tid & 15) << 2;     // float4 column within 64-wide chunk
  const int gr0 = rowBase + ra0, gr1 = rowBase + ra1;
  const int s0 = rowmap ? rowmap[gr0] : gr0;
  const int s1 = rowmap ? rowmap[gr1] : gr1;
  const float* aPtr0 = A + (size_t)s0 * 512 + ac;
  const float* aPtr1 = A + (size_t)s1 * 512 + ac;

  const int rb0 = tid >> 4;            // B rows handled: rb0 + 16*i
  const int cb  = (tid & 15) << 2;
  const float* bPtr[4];
  size_t bstep;                        // float offset per 64-wide K chunk
  if (transB) {
#pragma unroll
    for (int i = 0; i < 4; ++i)
      bPtr[i] = W + (size_t)(colBase + rb0 + 16 * i) * 512 + cb;   // row n, cols = k
    bstep = 64;
  } else {
#pragma unroll
    for (int i = 0; i < 4; ++i)
      bPtr[i] = W + (size_t)(rb0 + 16 * i) * 512 + colBase + cb;   // row k, cols = n
    bstep = (size_t)64 * 512;
  }

  v8f acc0 = {}; v8f acc1 = {};

  // ---- prologue: stage chunk 0 into registers ----
  float4 aR0 = *(const float4*)aPtr0;
  float4 aR1 = *(const float4*)aPtr1;
  float4 bR0 = *(const float4*)bPtr[0];
  float4 bR1 = *(const float4*)bPtr[1];
  float4 bR2 = *(const float4*)bPtr[2];
  float4 bR3 = *(const float4*)bPtr[3];

  for (int kc = 0; kc < 512; kc += 64) {
    __syncthreads();  // previous chunk's WMMAs done reading LDS
    // ---- commit staged registers to LDS ----
    *(float4*)(&As[ra0 * AS_LD + ac]) = aR0;
    *(float4*)(&As[ra1 * AS_LD + ac]) = aR1;
    if (transB) {
      // bR_i holds W[n, cb..cb+3] -> k = cb..cb+3, n = rb0+16i
#pragma unroll
      for (int i = 0; i < 4; ++i) {
        const float4 wv = (i == 0) ? bR0 : (i == 1) ? bR1 : (i == 2) ? bR2 : bR3;
        const int n = rb0 + 16 * i;
        Bs[((cb >> 1) + 0) * BS_GRP + n * 2 + 0] = wv.x;
        Bs[((cb >> 1) + 0) * BS_GRP + n * 2 + 1] = wv.y;
        Bs[((cb >> 1) + 1) * BS_GRP + n * 2 + 0] = wv.z;
        Bs[((cb >> 1) + 1) * BS_GRP + n * 2 + 1] = wv.w;
      }
    } else {
      // bR_i holds W[k, n..n+3] -> k = rb0+16i, n = cb..cb+3
#pragma unroll
      for (int i = 0; i < 4; ++i) {
        const float4 wv = (i == 0) ? bR0 : (i == 1) ? bR1 : (i == 2) ? bR2 : bR3;
        const int k = rb0 + 16 * i;
        const int g = (k >> 1) * BS_GRP + (k & 1);
        Bs[g + (cb + 0) * 2] = wv.x;
        Bs[g + (cb + 1) * 2] = wv.y;
        Bs[g + (cb + 2) * 2] = wv.z;
        Bs[g + (cb + 3) * 2] = wv.w;
      }
    }
    // ---- issue next chunk's global loads (overlap with WMMA below) ----
    if (kc + 64 < 512) {
      const size_t boff = (size_t)(kc / 64 + 1) * bstep;
      aR0 = *(const float4*)(aPtr0 + kc + 64);
      aR1 = *(const float4*)(aPtr1 + kc + 64);
      bR0 = *(const float4*)(bPtr[0] + boff);
      bR1 = *(const float4*)(bPtr[1] + boff);
      bR2 = *(const float4*)(bPtr[2] + boff);
      bR3 = *(const float4*)(bPtr[3] + boff);
    }
    __syncthreads();  // LDS tile visible to all waves
    // ---- 16 WMMA k-steps of 4 (two accumulators to break RAW chain) ----
    const int arow  = wm * 16 + (lane & 15);
    const int bcol  = wn * 16 + (lane & 15);
    const int khalf = lane >> 4;  // lanes 0-15: K0/K1, lanes 16-31: K2/K3
#pragma unroll
    for (int ks = 0; ks < 16; ks += 2) {
      int k0 = ks * 4;
      v2f a0 = *(const v2f*)(&As[arow * AS_LD + k0 + 2 * khalf]);
      v2f b0 = *(const v2f*)(&Bs[((k0 >> 1) + khalf) * BS_GRP + bcol * 2]);
      acc0 = __builtin_amdgcn_wmma_f32_16x16x4_f32(false, a0, false, b0,
                                                   (short)0, acc0, false, false);
      int k1 = k0 + 4;
      v2f a1 = *(const v2f*)(&As[arow * AS_LD + k1 + 2 * khalf]);
      v2f b1 = *(const v2f*)(&Bs[((k1 >> 1) + khalf) * BS_GRP + bcol * 2]);
      acc1 = __builtin_amdgcn_wmma_f32_16x16x4_f32(false, a1, false, b1,
                                                   (short)0, acc1, false, false);
    }
  }
  v8f acc = acc0 + acc1;
  // 16x16 f32 C/D layout: lanes 0-15 -> M = v, lanes 16-31 -> M = v+8; N = lane&15
  const int n    = colBase + wn * 16 + (lane & 15);
  const int mrow = rowBase + wm * 16 + ((lane >> 4) << 3);
  const float bv = bias ? bias[n] : 0.f;
#pragma unroll
  for (int v = 0; v < 8; ++v)
    D[(size_t)(mrow + v) * 512 + n] = acc[v] + bv;
}

// qg[b,d] = mean over p(4096),h(8) of q[b,p,h*64+d]
__global__ void qg_reduce(const float* __restrict__ q, float* __restrict__ qg) {
  const int b = blockIdx.x >> 6;
  const int d = blockIdx.x & 63;
  __shared__ float red[256];
  float acc = 0.f;
  for (int i = threadIdx.x; i < 32768; i += 256) {
    int p = i >> 3, h = i & 7;
    acc += q[((size_t)(b * 4096 + p)) * 512 + h * 64 + d];
  }
  red[threadIdx.x] = acc;
  __syncthreads();
  for (int s = 128; s; s >>= 1) {
    if (threadIdx.x < s) red[threadIdx.x] += red[threadIdx.x + s];
    __syncthreads();
  }
  if (threadIdx.x == 0) qg[b * 64 + d] = red[0] * (1.0f / 32768.0f);
}

// memory-slot attention -> rvec[b, c] = tanh(alpha) * (softmax(qg@mem^T/8)@mem)[c%64]
__global__ void mem_kernel(const float* __restrict__ qg, const float* __restrict__ mem,
                           const float* __restrict__ mem_alpha, float* __restrict__ rvec) {
  __shared__ float sc[2][16];
  __shared__ float at[2][16];
  const int t = threadIdx.x;
  if (t < 32) {
    int b = t >> 4, s = t & 15;
    float acc = 0.f;
    for (int d = 0; d < 64; ++d) acc += qg[b * 64 + d] * mem[s * 64 + d];
    sc[b][s] = acc * 0.125f;  // 1/sqrt(64)
  }
  __syncthreads();
  if (t < 2) {
    float m = -3.4e38f;
    for (int s = 0; s < 16; ++s) m = fmaxf(m, sc[t][s]);
    float e[16], den = 0.f;
    for (int s = 0; s < 16; ++s) { e[s] = __expf(sc[t][s] - m); den += e[s]; }
    for (int s = 0; s < 16; ++s) at[t][s] = e[s] / den;
  }
  __syncthreads();
  if (t < 128) {
    int b = t >> 6, d = t & 63;
    float r = 0.f;
    for (int s = 0; s < 16; ++s) r += at[b][s] * mem[s * 64 + d];
    float alpha = tanhf(mem_alpha[0]);
    for (int h = 0; h < 8; ++h) rvec[b * 512 + h * 64 + d] = alpha * r;
  }
}

// q += rvec (broadcast per batch); also emit the "now" output = hidden copy.
__global__ void augment_copy(float* __restrict__ q, const float* __restrict__ rvec,
                             const float* __restrict__ hidden, float* __restrict__ out_now) {
  size_t i = (size_t)blockIdx.x * 256 + threadIdx.x;  // 0..4194303
  int b = (int)(i >> 21);                              // bp/4096
  int c = (int)(i & 511);
  q[i] += rvec[b * 512 + c];
  out_now[i] = hidden[i];
}

// per-row (one wave per bp): sel[t] = qk[bp]·past[bp,t]; top-8 (jax tie rule), sorted asc.
__global__ void topk_kernel(const float* __restrict__ qk, const float* __restrict__ past,
                            int* __restrict__ rowmap) {
  const int wave = threadIdx.x >> 5;
  const int lane = threadIdx.x & 31;
  const int bp = blockIdx.x * 8 + wave;
  const float* qr = qk + (size_t)bp * 512;
  float qv[16];
#pragma unroll
  for (int i = 0; i < 16; ++i) qv[i] = qr[lane + 32 * i];
  float sel[16];
  for (int t = 0; t < 16; ++t) {
    const float* pr = past + ((size_t)bp * 16 + t) * 512;
    float acc = 0.f;
#pragma unroll
    for (int i = 0; i < 16; ++i) acc += qv[i] * pr[lane + 32 * i];
    for (int m = 16; m >= 1; m >>= 1) acc += __shfl_xor(acc, m, 32);
    sel[t] = acc;
  }
  unsigned used = 0;
  int chosen[8];
  for (int r = 0; r < 8; ++r) {  // strict '>' keeps earliest index on ties (matches top_k)
    float best = -3.4e38f; int bi = 0;
    for (int t = 0; t < 16; ++t)
      if (!((used >> t) & 1u) && sel[t] > best) { best = sel[t]; bi = t; }
    used |= 1u << bi;
    chosen[r] = bi;
  }
  for (int i = 1; i < 8; ++i) {  // ascending sort
    int v = chosen[i], j = i - 1;
    while (j >= 0 && chosen[j] > v) { chosen[j + 1] = chosen[j]; --j; }
    chosen[j + 1] = v;
  }
  if (lane < 8) rowmap[bp * 8 + lane] = bp * 16 + chosen[lane];
}

// one wave per (row, head): RoPE(q@8, k@0..8), 9-key softmax attention -> ctx
__global__ void attn_kernel(const float* __restrict__ q, const float* __restrict__ k_read,
                            const float* __restrict__ k_now, const float* __restrict__ v_read,
                            const float* __restrict__ v_now, float* __restrict__ ctx) {
  const int h = threadIdx.x >> 5;
  const int lane = threadIdx.x & 31;
  const int bp = blockIdx.x;
  const size_t qoff = (size_t)bp * 512 + h * 64;
  float q1 = q[qoff + lane], q2 = q[qoff + 32 + lane];
  float invf = __powf(10000.f, -(float)lane * (1.f / 32.f));
  float sq, cq; __sincosf(8.f * invf, &sq, &cq);
  float qr1 = q1 * cq - q2 * sq;
  float qr2 = q1 * sq + q2 * cq;
  float sc[9], v1[9], v2[9];
  for (int kk = 0; kk < 9; ++kk) {
    size_t off = (kk < 8) ? ((size_t)bp * 8 + kk) * 512 + h * 64 : qoff;
    const float* kp = (kk < 8) ? k_read : k_now;
    const float* vp = (kk < 8) ? v_read : v_now;
    float k1 = kp[off + lane], k2 = kp[off + 32 + lane];
    v1[kk] = vp[off + lane];
    v2[kk] = vp[off + 32 + lane];
    float sk, ck; __sincosf((float)kk * invf, &sk, &ck);
    float kr1 = k1 * ck - k2 * sk;
    float kr2 = k1 * sk + k2 * ck;
    float acc = qr1 * kr1 + qr2 * kr2;
    for (int m = 16; m >= 1; m >>= 1) acc += __shfl_xor(acc, m, 32);
    sc[kk] = acc * 0.125f;  // 1/sqrt(64)
  }
  float mx = sc[0];
  for (int kk = 1; kk < 9; ++kk) mx = fmaxf(mx, sc[kk]);
  float w[9], den = 0.f;
  for (int kk = 0; kk < 9; ++kk) { w[kk] = __expf(sc[kk] - mx); den += w[kk]; }
  float inv = 1.f / den;
  float c1 = 0.f, c2 = 0.f;
  for (int kk = 0; kk < 9; ++kk) { c1 += w[kk] * v1[kk]; c2 += w[kk] * v2[kk]; }
  ctx[qoff + lane] = c1 * inv;
  ctx[qoff + 32 + lane] = c2 * inv;
}

extern "C" void kernel_launch(void* const* d_in, const int* in_sizes, int n_in,
                              void* d_out, int out_size, void* d_ws, size_t ws_size,
                              hipStream_t stream) {
  const float* hidden    = (const float*)d_in[0];  // (2,4096,512)
  const float* past      = (const float*)d_in[1];  // (8192,16,512)
  const float* Wq        = (const float*)d_in[2];
  const float* Wk        = (const float*)d_in[3];
  const float* Wv        = (const float*)d_in[4];
  const float* Wout      = (const float*)d_in[5];
  const float* bout      = (const float*)d_in[6];
  const float* mem_slots = (const float*)d_in[7];  // (16,64)
  const float* mem_alpha = (const float*)d_in[8];  // (1,)
  (void)in_sizes; (void)n_in; (void)out_size; (void)ws_size;

  float* out_hs  = (float*)d_out;                 // 8192*512
  float* out_now = out_hs + (size_t)8192 * 512;   // 8192*512

  const size_t R = (size_t)8192 * 512;
  float* ws     = (float*)d_ws;
  float* q      = ws;                 // 8192x512 (augmented in place)
  float* k_now  = q + R;
  float* v_now  = k_now + R;
  float* qk     = v_now + R;
  float* ctx    = qk + R;
  float* k_read = ctx + R;            // 65536x512
  float* v_read = k_read + (size_t)65536 * 512;
  float* qg     = v_read + (size_t)65536 * 512;   // 128
  float* rvec   = qg + 128;                        // 1024
  int*   rowmap = (int*)(rvec + 1024);             // 65536

  const dim3 blk(256);
  const dim3 g8k(8192 / 32, 8);   // M=8192 tiles
  const dim3 g64k(65536 / 32, 8); // M=65536 tiles

  // Projections (D = A @ W^T)
  gemm512_wmma<<<g8k, blk, 0, stream>>>(hidden, Wq, q,     nullptr, nullptr, 1);
  gemm512_wmma<<<g8k, blk, 0, stream>>>(hidden, Wk, k_now, nullptr, nullptr, 1);
  gemm512_wmma<<<g8k, blk, 0, stream>>>(hidden, Wv, v_now, nullptr, nullptr, 1);
  // Memory-slot augmentation of q
  qg_reduce<<<dim3(128), blk, 0, stream>>>(q, qg);
  mem_kernel<<<dim3(1), dim3(128), 0, stream>>>(qg, mem_slots, mem_alpha, rvec);
  augment_copy<<<dim3(16384), blk, 0, stream>>>(q, rvec, hidden, out_now);
  // sel = q·(past@Wk^T) = (q@Wk)·past  -> qk = q @ Wk (no transpose)
  gemm512_wmma<<<g8k, blk, 0, stream>>>(q, Wk, qk, nullptr, nullptr, 0);
  topk_kernel<<<dim3(1024), blk, 0, stream>>>(qk, past, rowmap);
  // Project only the 8 selected past rows per token (gather via rowmap)
  gemm512_wmma<<<g64k, blk, 0, stream>>>(past, Wk, k_read, rowmap, nullptr, 1);
  gemm512_wmma<<<g64k, blk, 0, stream>>>(past, Wv, v_read, rowmap, nullptr, 1);
  // RoPE + 9-key attention
  attn_kernel<<<dim3(8192), blk, 0, stream>>>(q, k_read, k_now, v_read, v_now, ctx);
  // Output projection with bias
  gemm512_wmma<<<g8k, blk, 0, stream>>>(ctx, Wout, out_hs, nullptr, bout, 1);
}